// KWinners_31215822307921
// MI455X (gfx1250) — compile-verified
//
#include <hip/hip_runtime.h>
#include <hip/hip_bf16.h>
#include <stdint.h>

#define NCOLS 8192
#define NROWS 4096
#define KSEL  819
#define TPB   256
#define CHUNKS 8                      // float4 chunks per thread (8*4*256 = 8192)

// ---------------------------------------------------------------------------
// Kernel 1: precompute boost factors once (8192 exps total).
// boost[i] = exp((K/N - duty[i]) * boost_strength), boost_strength = 1.0
// ---------------------------------------------------------------------------
__global__ __launch_bounds__(TPB) void boost_kernel(const float* __restrict__ duty,
                                                    float* __restrict__ boost) {
    int i = blockIdx.x * TPB + threadIdx.x;
    if (i < NCOLS) {
        const float target_density = (float)KSEL / (float)NCOLS;
        boost[i] = expf((target_density - duty[i]) * 1.0f);
    }
}

// ---------------------------------------------------------------------------
// Kernel 2: one workgroup (8 wave32) per row.
//   1) async-DMA the 32KB row into LDS (CDNA5 GLOBAL_LOAD_ASYNC_TO_LDS_B128)
//   2) build monotone-sortable keys of x*boost in registers
//   3) reuse the (now dead) 32KB LDS as a 2048-bin histogram:
//      exact 11/11/10-bit MSB radix select of the K-th largest key
//   4) vectorized masked write-out (b128 stores)
// ---------------------------------------------------------------------------
__global__ __launch_bounds__(TPB) void kwta_kernel(const float* __restrict__ x,
                                                   const float* __restrict__ boost,
                                                   float* __restrict__ out) {
    __shared__ unsigned u_lds[NCOLS];   // 32KB: x staging, then histogram bins
    __shared__ unsigned scan[TPB];
    __shared__ unsigned s_prefix;
    __shared__ unsigned s_remaining;
    __shared__ unsigned s_tie;

    const int tid = threadIdx.x;
    const int row = blockIdx.x;
    const float* __restrict__ xr = x + (size_t)row * NCOLS;

    // ---- 1) Stage the row into LDS with async global->LDS DMA --------------
    // VDST VGPR carries the wave-relative LDS byte offset == low 32 bits of
    // the generic (flat) pointer to the __shared__ array.
    unsigned lds_base = (unsigned)(size_t)&u_lds[0];
#pragma unroll
    for (int c = 0; c < CHUNKS; ++c) {
        unsigned byte_off = (unsigned)((c * TPB + tid) * 16);
        unsigned lds_addr = lds_base + byte_off;
        asm volatile("global_load_async_to_lds_b128 %0, %1, %2"
                     :
                     : "v"(lds_addr), "v"(byte_off), "s"(xr)
                     : "memory");
    }
    asm volatile("s_wait_asynccnt 0" ::: "memory");
    __syncthreads();

    // ---- 2) Keys in registers (float4 granularity, matches DMA layout) -----
    float    xv[CHUNKS * 4];
    unsigned key[CHUNKS * 4];
    {
        const float4* __restrict__ ldsf4 = (const float4*)u_lds;
        const float4* __restrict__ bo4   = (const float4*)boost;
#pragma unroll
        for (int c = 0; c < CHUNKS; ++c) {
            float4 v4 = ldsf4[c * TPB + tid];       // ds_load_b128
            float4 b4 = bo4[c * TPB + tid];         // global_load_b128 (L2-hot)
            float vs[4] = {v4.x, v4.y, v4.z, v4.w};
            float bs[4] = {b4.x, b4.y, b4.z, b4.w};
#pragma unroll
            for (int q = 0; q < 4; ++q) {
                int e = c * 4 + q;
                xv[e] = vs[q];
                unsigned u = __float_as_uint(vs[q] * bs[q]);
                key[e] = (u & 0x80000000u) ? ~u : (u | 0x80000000u);
            }
        }
    }

    if (tid == 0) { s_prefix = 0u; s_remaining = (unsigned)KSEL; }
    __syncthreads();   // also fences last LDS read of x before hist reuse

    // ---- 3) 3-pass MSB radix select (11 + 11 + 10 bits) ---------------------
    const int      shifts[3] = {21, 10, 0};
    const unsigned dmsk[3]   = {0x7FFu, 0x7FFu, 0x3FFu};
    const unsigned himask[3] = {0u, 0xFFE00000u, 0xFFFFFC00u};
    const int      nbins[3]  = {2048, 2048, 1024};

    for (int p = 0; p < 3; ++p) {
        const int      sh = shifts[p];
        const unsigned dm = dmsk[p];
        const unsigned hm = himask[p];
        const int      nb = nbins[p];
        const int      ch = nb >> 8;            // bins per thread chunk (8/8/4)
        const unsigned sp = s_prefix;

        for (int i = tid; i < nb; i += TPB) u_lds[i] = 0u;
        __syncthreads();

#pragma unroll
        for (int e = 0; e < CHUNKS * 4; ++e) {
            unsigned k = key[e];
            if (((k ^ sp) & hm) == 0u) {
                atomicAdd(&u_lds[(k >> sh) & dm], 1u);   // ~conflict-free bins
            }
        }
        __syncthreads();

        // per-thread sum of its contiguous chunk of bins
        unsigned csum = 0u;
        for (int c = 0; c < ch; ++c) csum += u_lds[tid * ch + c];
        scan[tid] = csum;
        __syncthreads();

        // suffix-inclusive scan over the 256 chunk sums (Hillis-Steele)
        for (int off = 1; off < TPB; off <<= 1) {
            unsigned v = (tid + off < TPB) ? scan[tid + off] : 0u;
            __syncthreads();
            scan[tid] += v;
            __syncthreads();
        }

        const unsigned rem   = s_remaining;
        const unsigned incl  = scan[tid];        // keys with digit >= chunk base
        const unsigned above = incl - csum;      // keys in strictly higher chunks
        __syncthreads();                          // snapshot rem before the write
        if (above < rem && incl >= rem) {        // exactly one thread matches
            unsigned acc = above;
            for (int c = ch - 1; c >= 0; --c) {  // walk own bins, high -> low
                unsigned h = u_lds[tid * ch + c];
                if (acc + h >= rem) {
                    s_prefix    = sp | ((unsigned)(tid * ch + c) << sh);
                    s_remaining = rem - acc;
                    break;
                }
                acc += h;
            }
        }
        __syncthreads();
    }

    const unsigned thr = s_prefix;     // exact key of the K-th largest element
    const unsigned rem = s_remaining;  // ties at thr to keep (>= 1)
    if (tid == 0) s_tie = 0u;
    __syncthreads();

    // ---- 4) Masked, vectorized write-out -----------------------------------
    float4* __restrict__ orow4 = (float4*)(out + (size_t)row * NCOLS);
#pragma unroll
    for (int c = 0; c < CHUNKS; ++c) {
        float o[4];
#pragma unroll
        for (int q = 0; q < 4; ++q) {
            int e = c * 4 + q;
            float v = 0.0f;
            if (key[e] > thr) {
                v = xv[e];
            } else if (key[e] == thr) {
                unsigned t = atomicAdd(&s_tie, 1u);
                if (t < rem) v = xv[e];
            }
            o[q] = v;
        }
        float4 o4; o4.x = o[0]; o4.y = o[1]; o4.z = o[2]; o4.w = o[3];
        orow4[c * TPB + tid] = o4;                 // global_store_b128
    }
}

// ---------------------------------------------------------------------------
extern "C" void kernel_launch(void* const* d_in, const int* in_sizes, int n_in,
                              void* d_out, int out_size, void* d_ws, size_t ws_size,
                              hipStream_t stream) {
    const float* x    = (const float*)d_in[0];   // [4096, 8192] f32
    const float* duty = (const float*)d_in[1];   // [8192] f32
    float* out   = (float*)d_out;                // [4096, 8192] f32
    float* boost = (float*)d_ws;                 // 8192 f32 scratch

    boost_kernel<<<NCOLS / TPB, TPB, 0, stream>>>(duty, boost);
    kwta_kernel<<<NROWS, TPB, 0, stream>>>(x, boost, out);
}